// PoincareBallLinear_48387101557105
// MI455X (gfx1250) — compile-verified
//
#include <hip/hip_runtime.h>

// Problem constants (from reference): B=128, IN=1024, OUT=1024, CURV=1.0
#define PB_B   128
#define PB_IN  1024
#define PB_OUT 1024

typedef float v2f __attribute__((ext_vector_type(2)));
typedef float v4f __attribute__((ext_vector_type(4)));
typedef float v8f __attribute__((ext_vector_type(8)));

// ---------------------------------------------------------------------------
// Kernel 1: Poincare (Mobius) path in product form.
// Identity (c=1): representing r = (P-Q)/(P+Q), the Mobius step
//   r <- (r + b) / (1 + r*b)  becomes  P <- P*(1+b), Q <- Q*(1-b).
// Init P=Q=1 reproduces the reference's k==0 direct assignment exactly, and
// the bias Mobius-add is just one more factor. Inner loop is pure FMA — no
// transcendentals, no serial rcp chain.
//
// 2x2 register blocking: each thread owns (2 b) x (2 o); 4 LDS reads feed
// 4 elements. Block = 256 threads covering a 32x32 tile; K staged in 32-wide
// LDS chunks with stride 33 (w-row reads hit 16 distinct banks, pairs of
// lanes share addresses -> broadcast, no conflicts).
// Writes out = 0.05 * poincare (kernel 2 adds the 0.95*standard part).
// ---------------------------------------------------------------------------
#define KC       32
#define LSTRIDE  33

__global__ __launch_bounds__(256) void
poincare_mobius_kernel(const float* __restrict__ x,
                       const float* __restrict__ w,
                       const float* __restrict__ bias,
                       float* __restrict__ out)
{
    __shared__ float xs [32 * LSTRIDE];
    __shared__ float wsh[32 * LSTRIDE];

    const int tid = threadIdx.x;         // 0..255
    const int ol2 = tid & 15;            // o-pair index (0..15)
    const int bl2 = tid >> 4;            // b-pair index (0..15)
    const int o0  = blockIdx.x * 32;
    const int b0  = blockIdx.y * 32;

    // Staging role: each thread loads one float4 per array per chunk.
    const int ldr = tid >> 3;            // row 0..31
    const int ldc = (tid & 7) * 4;       // col 0,4,...,28

    float P00 = 1.0f, P01 = 1.0f, P10 = 1.0f, P11 = 1.0f;
    float Q00 = 1.0f, Q01 = 1.0f, Q10 = 1.0f, Q11 = 1.0f;

    const int xrow0 = 2 * bl2;
    const int orow0 = 2 * ol2;

    for (int k0 = 0; k0 < PB_IN; k0 += KC) {
        // Coalesced global loads into registers first (b128 per thread).
        v4f gx = *(const v4f*)(x + (size_t)(b0 + ldr) * PB_IN + k0 + ldc);
        v4f gw = *(const v4f*)(w + (size_t)(o0 + ldr) * PB_IN + k0 + ldc);

        __syncthreads();                 // previous chunk fully consumed
#pragma unroll
        for (int c = 0; c < 4; ++c) {
            xs [ldr * LSTRIDE + ldc + c] = gx[c];
            wsh[ldr * LSTRIDE + ldc + c] = gw[c];
        }
        __syncthreads();                 // tile visible to all

#pragma unroll 4
        for (int j = 0; j < KC; ++j) {
            const float xa = xs [(xrow0    ) * LSTRIDE + j];
            const float xb = xs [(xrow0 + 1) * LSTRIDE + j];
            const float wa = wsh[(orow0    ) * LSTRIDE + j];
            const float wb = wsh[(orow0 + 1) * LSTRIDE + j];

            const float t00 = xa * wa;
            const float t01 = xa * wb;
            const float t10 = xb * wa;
            const float t11 = xb * wb;

            P00 = __builtin_fmaf( t00, P00, P00);
            Q00 = __builtin_fmaf(-t00, Q00, Q00);
            P01 = __builtin_fmaf( t01, P01, P01);
            Q01 = __builtin_fmaf(-t01, Q01, Q01);
            P10 = __builtin_fmaf( t10, P10, P10);
            Q10 = __builtin_fmaf(-t10, Q10, Q10);
            P11 = __builtin_fmaf( t11, P11, P11);
            Q11 = __builtin_fmaf(-t11, Q11, Q11);
        }
    }

    // Fold the Mobius bias-add as one more product factor per o column.
    const float bva = bias[o0 + orow0];
    const float bvb = bias[o0 + orow0 + 1];
    P00 = __builtin_fmaf( bva, P00, P00);  Q00 = __builtin_fmaf(-bva, Q00, Q00);
    P10 = __builtin_fmaf( bva, P10, P10);  Q10 = __builtin_fmaf(-bva, Q10, Q10);
    P01 = __builtin_fmaf( bvb, P01, P01);  Q01 = __builtin_fmaf(-bvb, Q01, Q01);
    P11 = __builtin_fmaf( bvb, P11, P11);  Q11 = __builtin_fmaf(-bvb, Q11, Q11);

    // r = (P - Q) / (P + Q) — single v_rcp_f32 per element, off the hot loop.
    const float r00 = (P00 - Q00) * __builtin_amdgcn_rcpf(P00 + Q00);
    const float r01 = (P01 - Q01) * __builtin_amdgcn_rcpf(P01 + Q01);
    const float r10 = (P10 - Q10) * __builtin_amdgcn_rcpf(P10 + Q10);
    const float r11 = (P11 - Q11) * __builtin_amdgcn_rcpf(P11 + Q11);

    // Adjacent o pair -> float2 stores (8B aligned since o is even).
    const int row0 = b0 + xrow0;
    const int col  = o0 + orow0;
    v2f s0 = { 0.05f * r00, 0.05f * r01 };
    v2f s1 = { 0.05f * r10, 0.05f * r11 };
    *(v2f*)(out + (size_t)(row0    ) * PB_OUT + col) = s0;
    *(v2f*)(out + (size_t)(row0 + 1) * PB_OUT + col) = s1;
}

// ---------------------------------------------------------------------------
// Kernel 2: standard linear path via V_WMMA_F32_16X16X4_F32.
// One wave (32 threads) per 16x16 output tile; K-loop in steps of 8 with two
// independent accumulators (even/odd K groups) to break the serial WMMA
// accumulation chain — with only ~1 wave/SIMD resident, ILP is what hides
// the matrix-pipe latency.
// A (16x4 fp32): lanes 0-15 hold M=lane, VGPR0/1 = K,K+1; lanes 16-31 = K+2,K+3
//   -> contiguous float2 from x row (m0+lane&15) at col k + 2*(lane>>4).
// B (4x16 fp32): B[k][n] = weight[n][k]; mirrored layout -> same contiguous
//   float2 from weight row (n0+lane&15).
// D/C (16x16 f32): VGPR v, half h -> row m0 + v + 8h, col n0 + (lane&15).
// Accumulates into d_out on top of kernel 1's 0.05*poincare contribution.
// ---------------------------------------------------------------------------
__global__ __launch_bounds__(32) void
wmma_standard_kernel(const float* __restrict__ x,
                     const float* __restrict__ w,
                     const float* __restrict__ bias,
                     float* __restrict__ out)
{
    const int lane = threadIdx.x;      // 0..31 (wave32)
    const int half = lane >> 4;        // 0 or 1
    const int l16  = lane & 15;
    const int n0   = blockIdx.x * 16;
    const int m0   = blockIdx.y * 16;

    const float* __restrict__ arow = x + (size_t)(m0 + l16) * PB_IN + 2 * half;
    const float* __restrict__ brow = w + (size_t)(n0 + l16) * PB_IN + 2 * half;

    v8f acc0 = {};
    v8f acc1 = {};
#pragma unroll 4
    for (int k = 0; k < PB_IN; k += 8) {
        v2f a0 = *(const v2f*)(arow + k);
        v2f b0 = *(const v2f*)(brow + k);
        v2f a1 = *(const v2f*)(arow + k + 4);
        v2f b1 = *(const v2f*)(brow + k + 4);
        // 8 args: (neg_a, A, neg_b, B, c_mod, C, reuse_a, reuse_b)
        acc0 = __builtin_amdgcn_wmma_f32_16x16x4_f32(
            false, a0, false, b0, (short)0, acc0, false, false);
        acc1 = __builtin_amdgcn_wmma_f32_16x16x4_f32(
            false, a1, false, b1, (short)0, acc1, false, false);
    }

    const float bn = bias[n0 + l16];
#pragma unroll
    for (int v = 0; v < 8; ++v) {
        const int row = m0 + v + 8 * half;
        const int idx = row * PB_OUT + n0 + l16;
        out[idx] = __builtin_fmaf(0.95f, (acc0[v] + acc1[v]) + bn, out[idx]);
    }
}

// ---------------------------------------------------------------------------
extern "C" void kernel_launch(void* const* d_in, const int* in_sizes, int n_in,
                              void* d_out, int out_size, void* d_ws, size_t ws_size,
                              hipStream_t stream)
{
    (void)in_sizes; (void)n_in; (void)out_size; (void)d_ws; (void)ws_size;

    const float* x    = (const float*)d_in[0];   // [128, 1024]
    const float* w    = (const float*)d_in[1];   // [1024, 1024]
    const float* bias = (const float*)d_in[2];   // [1024]
    float*       out  = (float*)d_out;           // [128, 1024]

    // Kernel 1: writes out = 0.05 * poincare (full overwrite of d_out)
    dim3 g1(PB_OUT / 32, PB_B / 32);             // (32, 4)
    poincare_mobius_kernel<<<g1, 256, 0, stream>>>(x, w, bias, out);

    // Kernel 2: out += 0.95 * (x @ W^T + bias) via fp32 WMMA
    dim3 g2(PB_OUT / 16, PB_B / 16);             // (64, 8), one wave per tile
    wmma_standard_kernel<<<g2, 32, 0, stream>>>(x, w, bias, out);
}